// MFVIWordBasedConstituency_33904471834778
// MI455X (gfx1250) — compile-verified
//
#include <hip/hip_runtime.h>
#include <stdint.h>

#define BB 16
#define LL 128
#define MAX_ITER 3

typedef float        v2f   __attribute__((ext_vector_type(2)));
typedef float        v8f   __attribute__((ext_vector_type(8)));
typedef unsigned int u32x4 __attribute__((ext_vector_type(4)));
typedef int          i32x4 __attribute__((ext_vector_type(4)));
typedef int          i32x8 __attribute__((ext_vector_type(8)));

// One workgroup per (b,i): 256 threads = 8 wave32s, wave w owns j-rows [16w,16w+16).
//
// Masked matvec is decomposed as:
//   q'[j] = span[j] + mjf[j] * ( (SP_raw . v')[j] - sp[j,m]*v'[m] ),  m = max(i,j)
// with v'[k] = sigmoid(q[k]) * mkf[k].  The WMMA loop runs on the RAW tile:
// no per-element masking on the critical path.
__global__ __launch_bounds__(256)
void mfvi_kernel(const float* __restrict__ s_span,
                 const float* __restrict__ s_pair,
                 const unsigned char* __restrict__ mask,
                 float* __restrict__ out)
{
    __shared__ float sp_lds[LL * LL];   // 64KB raw s_pair tile (filled by TDM)
    __shared__ float q_lds[LL];
    __shared__ float v_lds[LL];         // v'[k] = sigmoid(q[k]) * mkf[k]
    __shared__ float span_lds[LL];
    __shared__ float mkf_lds[LL];       // m0[b,k] as 0/1 float
    __shared__ float mjf_lds[LL];       // mask[b,i,j]*m0[b,j] as 0/1 float

    const int bid  = blockIdx.x;        // bid = b*128 + i
    const int b    = bid >> 7;
    const int i    = bid & (LL - 1);
    const int tid  = threadIdx.x;
    const int lane = tid & 31;
    const int wave = tid >> 5;

    const size_t pair_base = (size_t)bid * (LL * LL);

    // ---- TDM: DMA the 64KB tile s_pair[b,i,:,:] global -> LDS (wave 0 only) ----
    if (wave == 0) {
        uint64_t gaddr  = (uint64_t)(uintptr_t)(s_pair + pair_base);
        uint32_t ldsoff = (uint32_t)(uintptr_t)(&sp_lds[0]);   // low 32b of flat LDS addr = offset
        u32x4 g0;
        g0[0] = 1u;                                            // count=1, user mode
        g0[1] = ldsoff;                                        // lds_addr
        g0[2] = (uint32_t)gaddr;                               // global_addr[31:0]
        g0[3] = (uint32_t)((gaddr >> 32) & 0x1FFFFFFu) | (2u << 30); // addr[56:32] | type=2
        i32x8 g1;
        g1[0] = 0x00020000;            // workgroup_mask=0, data_size=2 (4B)
        g1[1] = (int)(128u << 16);     // tensor_dim0 = 128
        g1[2] = (int)(128u << 16);     // tensor_dim1 = 128
        g1[3] = (int)(128u << 16);     // tile_dim0   = 128
        g1[4] = 128;                   // tile_dim1 = 128, tile_dim2 = 0
        g1[5] = 128;                   // tensor_dim0_stride = 128
        g1[6] = 0;
        g1[7] = 0;
        i32x4 gz4 = {0, 0, 0, 0};
#if __has_include(<hip/amd_detail/amd_gfx1250_TDM.h>)
        i32x8 gz8 = {0, 0, 0, 0, 0, 0, 0, 0};
        __builtin_amdgcn_tensor_load_to_lds(g0, g1, gz4, gz4, gz8, 0);   // clang-23 6-arg form
#else
        __builtin_amdgcn_tensor_load_to_lds(g0, g1, gz4, gz4, 0);        // ROCm 7.2 5-arg form
#endif
        __builtin_amdgcn_s_wait_tensorcnt(0);
    }

    // ---- init q = s_span row; per-k and per-j mask factors ----
    if (tid < LL) {
        float s = s_span[(size_t)bid * LL + tid];
        span_lds[tid] = s;
        q_lds[tid]    = s;
        float m0t = mask[(size_t)b * (LL * LL) + tid] ? 1.0f : 0.0f;     // m0[b,t]
        mkf_lds[tid]  = m0t;
        mjf_lds[tid]  = mask[(size_t)bid * LL + tid] ? m0t : 0.0f;       // mask[b,i,t]*m0[b,t]
    }
    __syncthreads();

    // 16x4 f32 A layout: lanes 0-15 hold M=lane, VGPR0/1 = K+0/K+1; lanes 16-31: K+2/K+3.
    const int jb   = wave << 4;
    const int j    = jb + (lane & 15);
    const int koff = (lane >> 4) << 1;

    // ---- 3 fused MFVI iterations ----
    for (int it = 0; it < MAX_ITER; ++it) {
        if (tid < LL) {
            float x = q_lds[tid];
            v_lds[tid] = mkf_lds[tid] / (1.0f + __expf(-x));   // v' = sigmoid(q)*mkf
        }
        __syncthreads();   // v' ready

        // y(16x16, all columns identical) = sum_c sp[jtile, 4c..](16x4) * B_c(4x16, v' replicated)
        v8f acc = {};
#pragma unroll
        for (int c = 0; c < 32; ++c) {
            v2f a  = *(const v2f*)&sp_lds[j * LL + (c << 2) + koff];
            v2f bb = *(const v2f*)&v_lds[(c << 2) + koff];
            acc = __builtin_amdgcn_wmma_f32_16x16x4_f32(
                      false, a, false, bb, (short)0, acc, false, false);
        }

        // column N=0 of D: lane 0 VGPR r -> y[jb+r]; lane 16 VGPR r -> y[jb+8+r].
        // Apply the single-k correction (k = max(i,j)) and the per-j mask factor.
        if ((lane & 15) == 0) {
            const int base = jb + ((lane >> 4) << 3);
#pragma unroll
            for (int r = 0; r < 8; ++r) {
                const int jj = base + r;
                const int m  = (i > jj) ? i : jj;                  // max(i,j)
                const float corr = sp_lds[jj * LL + m] * v_lds[m];
                q_lds[jj] = span_lds[jj] + mjf_lds[jj] * (acc[r] - corr);
            }
        }
        __syncthreads();   // q ready for next iteration's sigmoid
    }

    if (tid < LL)
        out[(size_t)bid * LL + tid] = q_lds[tid];
}

extern "C" void kernel_launch(void* const* d_in, const int* in_sizes, int n_in,
                              void* d_out, int out_size, void* d_ws, size_t ws_size,
                              hipStream_t stream) {
    const float*         s_span = (const float*)d_in[0];
    const float*         s_pair = (const float*)d_in[1];
    const unsigned char* mask   = (const unsigned char*)d_in[2];
    float*               out    = (float*)d_out;
    (void)in_sizes; (void)n_in; (void)out_size; (void)d_ws; (void)ws_size;

    dim3 grid(BB * LL);   // 2048 workgroups, one per (b,i)
    dim3 block(256);      // 8 wave32s
    hipLaunchKernelGGL(mfvi_kernel, grid, block, 0, stream, s_span, s_pair, mask, out);
}